// GuidedFilter_82446192214517
// MI455X (gfx1250) — compile-verified
//
#include <hip/hip_runtime.h>
#include <cstdint>

#define RAD    15
#define KW     31          // 2*RAD+1
#define TILE   64
#define REG    94          // TILE + 2*RAD ; also the LDS row stride for staged tiles
#define HSTR   65          // padded LDS row stride for horizontal sums (bank-spread)
#define IMG_H  1024
#define IMG_W  1024
#define EPS    0.001f

typedef unsigned int u32x4 __attribute__((ext_vector_type(4)));
typedef int          i32x4 __attribute__((ext_vector_type(4)));
typedef int          i32x8 __attribute__((ext_vector_type(8)));

// ---- CDNA5 per-lane async global -> LDS copy (ASYNCcnt-tracked) ----
static __device__ __forceinline__ void async_ld_f32(const void* lds_dst, const float* gsrc) {
  // Low 32 bits of a generic pointer into LDS == byte offset within the WG's LDS.
  unsigned lds_off = (unsigned)(uintptr_t)lds_dst;
  asm volatile("global_load_async_to_lds_b32 %0, %1, off"
               :: "v"(lds_off), "v"(gsrc)
               : "memory");
}
static __device__ __forceinline__ void wait_async_lds() {
  asm volatile("s_wait_asynccnt 0" ::: "memory");
}

// ---- Tensor DMA descriptor builder (2D tile, fp32, row stride IMG_W) ----
// tensor_dim* are relative to the tile's global start address; TDM zero-fills
// loads beyond a tensor_dim and drops stores beyond it.
static __device__ __forceinline__ void tdm_desc_2d(u32x4& g0, i32x8& g1,
                                                   unsigned lds_byte, uint64_t ga,
                                                   unsigned tdim0, unsigned tdim1,
                                                   unsigned cols, unsigned rows) {
  g0[0] = 1u;                                          // count=1, user descriptor
  g0[1] = lds_byte;                                    // lds_addr (bytes)
  g0[2] = (unsigned)ga;                                // global_addr[31:0]
  g0[3] = (unsigned)(ga >> 32) | 0x80000000u;          // global_addr[56:32] | type=2

  g1[0] = (int)(2u << 16);                             // data_size = 4 bytes
  g1[1] = (int)((tdim0 & 0xFFFFu) << 16);              // tensor_dim0[15:0]
  g1[2] = (int)((tdim0 >> 16) | ((tdim1 & 0xFFFFu) << 16));
  g1[3] = (int)((tdim1 >> 16) | (cols << 16));         // tile_dim0
  g1[4] = (int)rows;                                   // tile_dim1 ; tile_dim2 = 0
  g1[5] = (int)(unsigned)IMG_W;                        // tensor_dim0_stride
  g1[6] = 0;
  g1[7] = 0;
}

static __device__ __forceinline__ void tdm_load_2d(unsigned lds_byte, const float* gbase,
                                                   int gy, int gx, int rows) {
  int avail_y = IMG_H - gy; if (avail_y < 0) avail_y = 0;
  int gy_c    = (gy < IMG_H) ? gy : (IMG_H - 1);       // keep address sane when fully OOB
  uint64_t ga = (uint64_t)(uintptr_t)(gbase + (size_t)gy_c * IMG_W + gx);
  u32x4 g0; i32x8 g1;
  tdm_desc_2d(g0, g1, lds_byte, ga, (unsigned)(IMG_W - gx), (unsigned)avail_y,
              (unsigned)REG, (unsigned)rows);
  i32x4 gz4 = {0, 0, 0, 0};
  i32x8 gz8 = {0, 0, 0, 0, 0, 0, 0, 0};
  __builtin_amdgcn_tensor_load_to_lds(g0, g1, gz4, gz4, gz8, 0);
}

static __device__ __forceinline__ void tdm_store_2d(unsigned lds_byte, float* gbase,
                                                    int gy, int gx, int rows) {
  uint64_t ga = (uint64_t)(uintptr_t)(gbase + (size_t)gy * IMG_W + gx);
  u32x4 g0; i32x8 g1;
  tdm_desc_2d(g0, g1, lds_byte, ga, (unsigned)(IMG_W - gx), (unsigned)(IMG_H - gy),
              (unsigned)TILE, (unsigned)rows);
  i32x4 gz4 = {0, 0, 0, 0};
  i32x8 gz8 = {0, 0, 0, 0, 0, 0, 0, 0};
  __builtin_amdgcn_tensor_store_from_lds(g0, g1, gz4, gz4, gz8, 0);
}

// Stage a REG x REG halo region of two planes into LDS (row stride REG),
// zero-filled outside the image (reference = zero-padded conv, fixed 1/k^2).
static __device__ __forceinline__ void stage2(float* s0, float* s1,
                                              const float* g0, const float* g1p,
                                              int ty0, int tx0, int tid) {
  if (tx0 > 0) {
    // ---- TDM path: left halo is in-image; top handled by shift+pre-zero,
    //      right/bottom zero-filled by TDM OOB handling. ----
    int gx0 = tx0 - RAD;
    int gy0 = ty0 - RAD;
    int ry0 = 0;
    if (gy0 < 0) { ry0 = -gy0; gy0 = 0; }
    for (int i = tid; i < ry0 * REG; i += 256) {       // pre-zero top halo band
      s0[i] = 0.f;
      s1[i] = 0.f;
    }
    const int rows = REG - ry0;
    const int w    = __builtin_amdgcn_readfirstlane((int)(tid >> 5)); // wave id 0..7
    const int rbeg = (rows * w) >> 3;
    const int rend = (rows * (w + 1)) >> 3;
    const int nr   = rend - rbeg;                      // >= 9 always
    const unsigned lofs = (unsigned)((ry0 + rbeg) * REG) * 4u;
    tdm_load_2d((unsigned)(uintptr_t)(const void*)s0 + lofs, g0,  gy0 + rbeg, gx0, nr);
    tdm_load_2d((unsigned)(uintptr_t)(const void*)s1 + lofs, g1p, gy0 + rbeg, gx0, nr);
    __builtin_amdgcn_s_wait_tensorcnt(0);
    __syncthreads();
  } else {
    // ---- per-lane async fallback for left-edge tiles (tile_dim0 would change) ----
    for (int i = tid; i < REG * REG; i += 256) {
      int ry = i / REG;
      int rx = i - ry * REG;
      int gy = ty0 - RAD + ry;
      int gx = tx0 - RAD + rx;
      if ((unsigned)gy < (unsigned)IMG_H && (unsigned)gx < (unsigned)IMG_W) {
        size_t go = (size_t)gy * IMG_W + gx;
        async_ld_f32(&s0[i], g0 + go);
        async_ld_f32(&s1[i], g1p + go);
      } else {
        s0[i] = 0.f;
        s1[i] = 0.f;
      }
    }
    wait_async_lds();
    __syncthreads();
  }
}

// ---------------- Kernel 1: box stats -> a, b ----------------
__global__ __launch_bounds__(256)
void gf_stats(const float* __restrict__ I, const float* __restrict__ P,
              float* __restrict__ Aout, float* __restrict__ Bout) {
  __shared__ float sI[REG * REG];
  __shared__ float sP[REG * REG];
  __shared__ float h [REG * HSTR];

  const int tid = threadIdx.x;
  const int tx0 = blockIdx.x * TILE;
  const int ty0 = blockIdx.y * TILE;
  const size_t poff = (size_t)blockIdx.z * (size_t)IMG_H * IMG_W;
  const float* gI = I + poff;
  const float* gP = P + poff;

  stage2(sI, sP, gI, gP, ty0, tx0, tid);

  float acc[4][16];
  const int col = tid & (TILE - 1);
  const int r0  = (tid >> 6) * 16;

#pragma unroll
  for (int q = 0; q < 4; ++q) {
    // ---- horizontal running sum: 2 workers per region row ----
    if (tid < 2 * REG) {
      const int row = tid >> 1;
      const int c0  = (tid & 1) * (TILE / 2);
      const int rb  = row * REG;
      float s = 0.f;
#pragma unroll
      for (int j = 0; j < KW; ++j) {
        const int ix = rb + c0 + j;
        const float v = (q == 0) ? sI[ix]
                      : (q == 1) ? sP[ix]
                      : (q == 2) ? sI[ix] * sP[ix]
                                 : sI[ix] * sI[ix];
        s += v;
      }
      h[row * HSTR + c0] = s;
      for (int c = c0 + 1; c < c0 + TILE / 2; ++c) {
        const int ia = rb + c + KW - 1;
        const int ir = rb + c - 1;
        const float va = (q == 0) ? sI[ia] : (q == 1) ? sP[ia]
                       : (q == 2) ? sI[ia] * sP[ia] : sI[ia] * sI[ia];
        const float vr = (q == 0) ? sI[ir] : (q == 1) ? sP[ir]
                       : (q == 2) ? sI[ir] * sP[ir] : sI[ir] * sI[ir];
        s += va - vr;
        h[row * HSTR + c] = s;
      }
    }
    __syncthreads();
    // ---- vertical running sum: thread = (col, 16-row chunk) ----
    {
      float s = 0.f;
#pragma unroll
      for (int j = 0; j < KW; ++j) s += h[(r0 + j) * HSTR + col];
      acc[q][0] = s;
#pragma unroll
      for (int k = 1; k < 16; ++k) {
        s += h[(r0 + k + KW - 1) * HSTR + col] - h[(r0 + k - 1) * HSTR + col];
        acc[q][k] = s;
      }
    }
    __syncthreads();
  }

  // ---- a,b into (now dead) staging LDS, then DMA out via TDM ----
  const float inv = 1.0f / (float)(KW * KW);
#pragma unroll
  for (int k = 0; k < 16; ++k) {
    const float mI  = acc[0][k] * inv;
    const float mP  = acc[1][k] * inv;
    const float mIp = acc[2][k] * inv;
    const float mII = acc[3][k] * inv;
    const float var = mII - mI * mI;
    const float cov = mIp - mI * mP;
    const float a   = cov / (var + EPS);
    const float b   = mP - a * mI;
    sI[(r0 + k) * TILE + col] = a;
    sP[(r0 + k) * TILE + col] = b;
  }
  __syncthreads();
  {
    const int w  = __builtin_amdgcn_readfirstlane((int)(tid >> 5)); // wave id 0..7
    const int rb = w * (TILE / 8);                                  // 8 rows per wave
    const unsigned lofs = (unsigned)(rb * TILE) * 4u;
    tdm_store_2d((unsigned)(uintptr_t)(const void*)sI + lofs, Aout + poff, ty0 + rb, tx0, TILE / 8);
    tdm_store_2d((unsigned)(uintptr_t)(const void*)sP + lofs, Bout + poff, ty0 + rb, tx0, TILE / 8);
    __builtin_amdgcn_s_wait_tensorcnt(0);
  }
}

// ---------------- Kernel 2: box(a), box(b) -> q = mean_a*I + mean_b ----------------
__global__ __launch_bounds__(256)
void gf_apply(const float* __restrict__ Ain, const float* __restrict__ Bin,
              const float* __restrict__ I, float* __restrict__ Q) {
  __shared__ float sA[REG * REG];
  __shared__ float sB[REG * REG];
  __shared__ float h [REG * HSTR];

  const int tid = threadIdx.x;
  const int tx0 = blockIdx.x * TILE;
  const int ty0 = blockIdx.y * TILE;
  const size_t poff = (size_t)blockIdx.z * (size_t)IMG_H * IMG_W;

  stage2(sA, sB, Ain + poff, Bin + poff, ty0, tx0, tid);

  float acc[2][16];
  const int col = tid & (TILE - 1);
  const int r0  = (tid >> 6) * 16;

#pragma unroll
  for (int q = 0; q < 2; ++q) {
    if (tid < 2 * REG) {
      const int row = tid >> 1;
      const int c0  = (tid & 1) * (TILE / 2);
      const int rb  = row * REG;
      const float* src = (q == 0) ? sA : sB;
      float s = 0.f;
#pragma unroll
      for (int j = 0; j < KW; ++j) s += src[rb + c0 + j];
      h[row * HSTR + c0] = s;
      for (int c = c0 + 1; c < c0 + TILE / 2; ++c) {
        s += src[rb + c + KW - 1] - src[rb + c - 1];
        h[row * HSTR + c] = s;
      }
    }
    __syncthreads();
    {
      float s = 0.f;
#pragma unroll
      for (int j = 0; j < KW; ++j) s += h[(r0 + j) * HSTR + col];
      acc[q][0] = s;
#pragma unroll
      for (int k = 1; k < 16; ++k) {
        s += h[(r0 + k + KW - 1) * HSTR + col] - h[(r0 + k - 1) * HSTR + col];
        acc[q][k] = s;
      }
    }
    __syncthreads();
  }

  // ---- q into (now dead) staging LDS, then DMA out via TDM ----
  const float inv = 1.0f / (float)(KW * KW);
#pragma unroll
  for (int k = 0; k < 16; ++k) {
    const float mA = acc[0][k] * inv;
    const float mB = acc[1][k] * inv;
    const size_t o = poff + (size_t)(ty0 + r0 + k) * IMG_W + (tx0 + col);
    sA[(r0 + k) * TILE + col] = mA * I[o] + mB;
  }
  __syncthreads();
  {
    const int w  = __builtin_amdgcn_readfirstlane((int)(tid >> 5)); // wave id 0..7
    const int rb = w * (TILE / 8);                                  // 8 rows per wave
    const unsigned lofs = (unsigned)(rb * TILE) * 4u;
    tdm_store_2d((unsigned)(uintptr_t)(const void*)sA + lofs, Q + poff, ty0 + rb, tx0, TILE / 8);
    __builtin_amdgcn_s_wait_tensorcnt(0);
  }
}

extern "C" void kernel_launch(void* const* d_in, const int* in_sizes, int n_in,
                              void* d_out, int out_size, void* d_ws, size_t ws_size,
                              hipStream_t stream) {
  (void)n_in; (void)out_size; (void)ws_size;
  const float* I = (const float*)d_in[0];
  const float* P = (const float*)d_in[1];
  float* Q = (float*)d_out;

  const size_t N = (size_t)in_sizes[0];                 // B*C*H*W
  float* Aws = (float*)d_ws;                            // needs 2*N*4 bytes of ws
  float* Bws = Aws + N;
  const int planes = (int)(N / ((size_t)IMG_H * IMG_W));

  dim3 grid(IMG_W / TILE, IMG_H / TILE, planes);
  dim3 block(256);
  hipLaunchKernelGGL(gf_stats, grid, block, 0, stream, I, P, Aws, Bws);
  hipLaunchKernelGGL(gf_apply, grid, block, 0, stream, Aws, Bws, I, Q);
}